// BaseIterativeNet_29180007809598
// MI455X (gfx1250) — compile-verified
//
#include <hip/hip_runtime.h>
#include <hip/hip_bf16.h>

// Problem constants (fixed by setup_inputs)
#define L_DIM   512
#define LP_DIM  64
#define K_DIM   17
#define KP_DIM  20          // K padded to multiple of 4 for wmma 16x16x4
#define CH_DIM  51          // 3*K
#define FDIM    56
#define FEAT    (FDIM*FDIM) // 3136
#define FEAT_ELEMS (L_DIM * CH_DIM * FEAT)   // 81,887,232
#define EPS_AREA 1e-06f
#define EPS_BCE  1e-07f

typedef float v2f __attribute__((ext_vector_type(2)));
typedef float v4f __attribute__((ext_vector_type(4)));
typedef float v8f __attribute__((ext_vector_type(8)));

// ---------------------------------------------------------------------------
// Kernel A: broadcast pred_poses (L,3K) -> (L,3K,56,56). Pure streaming write.
// One block per (l,c) row: 3136 floats = 784 x 16B. Non-temporal B128 stores.
// ---------------------------------------------------------------------------
__global__ void bcast_kernel(const float* __restrict__ pred, float* __restrict__ out) {
    const int row = blockIdx.x;             // 0 .. L*3K-1
    const float v = pred[row];
    v4f v4 = { v, v, v, v };
    v4f* dst = reinterpret_cast<v4f*>(out + (size_t)row * FEAT);
    for (int i = threadIdx.x; i < FEAT / 4; i += blockDim.x) {
        __builtin_nontemporal_store(v4, dst + i);
    }
}

// ---------------------------------------------------------------------------
// Kernel B: loss. One workgroup per 16-row L tile (32 WGs). 4 waves/WG, each
// wave owns one 16-col Lp tile. BCE part via V_WMMA_F32_16X16X4_F32:
//   bce_sum = A(Lx17) @ tv^T(17xLp) + B(Lx17) @ (1-tv)^T
// Gaussian/exp part computed per-lane against the C-tile layout.
// Per-block partial written to d_ws (deterministic final reduce in kernel C).
// ---------------------------------------------------------------------------
__global__ void loss_kernel(const float* __restrict__ pred_poses,   // (L,K,3)
                            const float* __restrict__ target_poses, // (Lp,K,3)
                            const float* __restrict__ pred_fb,      // (L,)
                            const float* __restrict__ ious,         // (L,Lp)
                            const float* __restrict__ target_areas, // (Lp,)
                            float* __restrict__ partials) {
    __shared__ float Alds[16][KP_DIM];       // -log(pv), zero padded
    __shared__ float Blds[16][KP_DIM];       // -log(1-pv), zero padded
    __shared__ float Pxy [16][2 * K_DIM];    // pred x,y interleaved
    __shared__ float Tvl [LP_DIM][KP_DIM];   // target visibility, zero padded
    __shared__ float Txy [LP_DIM][2 * K_DIM];
    __shared__ float Invd[LP_DIM];
    __shared__ float red [128];

    const int tid    = threadIdx.x;
    const int l_base = blockIdx.x * 16;

    // ---- stage operands into LDS ----
    for (int idx = tid; idx < 16 * KP_DIM; idx += 128) {
        const int r = idx / KP_DIM, k = idx % KP_DIM;
        float a = 0.0f, b = 0.0f;
        if (k < K_DIM) {
            float pv = pred_poses[(l_base + r) * CH_DIM + k * 3 + 2];
            pv = fminf(fmaxf(pv, EPS_BCE), 1.0f - EPS_BCE);
            a = -__logf(pv);
            b = -__logf(1.0f - pv);
        }
        Alds[r][k] = a;
        Blds[r][k] = b;
    }
    for (int idx = tid; idx < 16 * K_DIM; idx += 128) {
        const int r = idx / K_DIM, k = idx % K_DIM;
        Pxy[r][2 * k]     = pred_poses[(l_base + r) * CH_DIM + k * 3 + 0];
        Pxy[r][2 * k + 1] = pred_poses[(l_base + r) * CH_DIM + k * 3 + 1];
    }
    for (int idx = tid; idx < LP_DIM * KP_DIM; idx += 128) {
        const int lp = idx / KP_DIM, k = idx % KP_DIM;
        Tvl[lp][k] = (k < K_DIM) ? target_poses[lp * CH_DIM + k * 3 + 2] : 0.0f;
    }
    for (int idx = tid; idx < LP_DIM * K_DIM; idx += 128) {
        const int lp = idx / K_DIM, k = idx % K_DIM;
        Txy[lp][2 * k]     = target_poses[lp * CH_DIM + k * 3 + 0];
        Txy[lp][2 * k + 1] = target_poses[lp * CH_DIM + k * 3 + 1];
    }
    for (int idx = tid; idx < LP_DIM; idx += 128) {
        Invd[idx] = 1.0f / (2.0f * target_areas[idx] + EPS_AREA);
    }
    __syncthreads();

    // ---- WMMA: bce_sum tile (16x16 f32) ----
    const int lane    = tid & 31;
    const int wv      = tid >> 5;        // 4 waves
    const int lp_base = wv * 16;
    const int r       = lane & 15;
    const int half    = lane >> 4;       // 0: K=k0,k0+1  1: K=k0+2,k0+3

    v8f acc = {};
#pragma unroll
    for (int ks = 0; ks < 5; ++ks) {
        const int k0 = ks * 4 + 2 * half;
        v2f a, b, a2, b2;
        a.x  = Alds[r][k0];
        a.y  = Alds[r][k0 + 1];
        b.x  = Tvl[lp_base + r][k0];
        b.y  = Tvl[lp_base + r][k0 + 1];
        a2.x = Blds[r][k0];
        a2.y = Blds[r][k0 + 1];
        b2.x = 1.0f - b.x;               // (1 - tv); A-side padding is zero
        b2.y = 1.0f - b.y;
        acc = __builtin_amdgcn_wmma_f32_16x16x4_f32(false, a,  false, b,
                                                    (short)0, acc, false, false);
        acc = __builtin_amdgcn_wmma_f32_16x16x4_f32(false, a2, false, b2,
                                                    (short)0, acc, false, false);
    }

    // ---- Gaussian feedback term against the same C-tile layout ----
    // Lane owns N = lane&15, rows M = half*8 + v, v=0..7 (ISA 16x16 f32 C/D map)
    const int lp    = lp_base + (lane & 15);
    const int m_off = half * 8;
    const float invd = Invd[lp];

    float s[8];
#pragma unroll
    for (int v = 0; v < 8; ++v) s[v] = 0.0f;

#pragma unroll
    for (int k = 0; k < K_DIM; ++k) {
        const float tx = Txy[lp][2 * k];
        const float ty = Txy[lp][2 * k + 1];
#pragma unroll
        for (int v = 0; v < 8; ++v) {
            const float dx = Pxy[m_off + v][2 * k]     - tx;
            const float dy = Pxy[m_off + v][2 * k + 1] - ty;
            s[v] += __expf(-(dx * dx + dy * dy) * invd);
        }
    }

    float part = 0.0f;
#pragma unroll
    for (int v = 0; v < 8; ++v) {
        const int l = l_base + m_off + v;
        const float tf  = s[v] * (1.0f / (float)K_DIM);
        const float d   = pred_fb[l] - tf;
        const float bce = acc[v] * (1.0f / (float)K_DIM);
        part += ious[l * LP_DIM + lp] * (d * d + bce);
    }

    // ---- deterministic in-block reduce ----
    red[tid] = part;
    __syncthreads();
    if (tid == 0) {
        float t = 0.0f;
        for (int i = 0; i < 128; ++i) t += red[i];
        partials[blockIdx.x] = t;
    }
}

// ---------------------------------------------------------------------------
// Kernel C: deterministic fixed-order final sum of 32 block partials.
// ---------------------------------------------------------------------------
__global__ void finalize_kernel(const float* __restrict__ partials,
                                float* __restrict__ out_loss) {
    if (threadIdx.x == 0 && blockIdx.x == 0) {
        float t = 0.0f;
        for (int i = 0; i < L_DIM / 16; ++i) t += partials[i];
        *out_loss = t;
    }
}

extern "C" void kernel_launch(void* const* d_in, const int* in_sizes, int n_in,
                              void* d_out, int out_size, void* d_ws, size_t ws_size,
                              hipStream_t stream) {
    const float* pred_poses   = (const float*)d_in[0]; // (512,17,3)
    const float* target_poses = (const float*)d_in[1]; // (64,17,3)
    const float* pred_fb      = (const float*)d_in[2]; // (512,)
    const float* ious         = (const float*)d_in[3]; // (512,64)
    const float* target_areas = (const float*)d_in[4]; // (64,)
    // d_in[5] = features_dim (56), fixed by the problem instance.

    float* out      = (float*)d_out;                 // features (81,887,232) + loss (1)
    float* partials = (float*)d_ws;                  // 32 floats of scratch

    // Kernel A: 512*51 = 26112 rows, one block each, streaming NT B128 stores.
    bcast_kernel<<<L_DIM * CH_DIM, 256, 0, stream>>>(pred_poses, out);

    // Kernel B: 32 workgroups (one per 16-row L tile), 128 threads (4 waves).
    loss_kernel<<<L_DIM / 16, 128, 0, stream>>>(pred_poses, target_poses, pred_fb,
                                                ious, target_areas, partials);

    // Kernel C: fixed-order sum -> loss scalar at the end of d_out.
    finalize_kernel<<<1, 32, 0, stream>>>(partials, out + (size_t)FEAT_ELEMS);
}